// TreeDecoder_72602127172122
// MI455X (gfx1250) — compile-verified
//
#include <hip/hip_runtime.h>
#include <hip/hip_bf16.h>

#define NN   256
#define HH   450
#define LATD 56
#define VV   780
#define KP   452   // H rounded up to multiple of 4 (WMMA K step)
#define LW   452   // row stride of labH / A2 (K-padded)
#define N1P  464   // 450 -> 29 tiles of 16
#define N2P  784   // 780 -> 49 tiles of 16

typedef float v2f __attribute__((ext_vector_type(2)));
typedef float v8f __attribute__((ext_vector_type(8)));

// ---------------------------------------------------------------- utilities
__global__ void zero_kernel(float* p, long n) {
  long i = (long)blockIdx.x * blockDim.x + threadIdx.x;
  long stride = (long)gridDim.x * blockDim.x;
  for (; i < n; i += stride) p[i] = 0.f;
}

__global__ void vid_kernel(const float* nf, int* vid) {
  int a = blockIdx.x;
  for (int i = threadIdx.x; i < VV; i += blockDim.x)
    if (nf[a * VV + i] > 0.5f) vid[a] = i;
}

// per-node constants: fold one-hot columns + biases + latent terms
__global__ __launch_bounds__(512) void pernode_kernel(
    const int* vid, const float* latent,
    const float* Wz_w, const float* Wz_b,
    const float* Wr_w, const float* Wr_b,
    const float* Wh_w, const float* Wh_b,
    const float* Wd12_w, const float* Wd12_b,
    float* dz, float* dr1, float* dh, float* d12c) {
  int a = blockIdx.x, i = threadIdx.x;
  if (i >= HH) return;
  int v = vid[a];
  dz[a * HH + i]  = Wz_w[i * (VV + HH) + v] + Wz_b[i];
  dr1[a * HH + i] = Wr_w[i * VV + v] + Wr_b[i];
  dh[a * HH + i]  = Wh_w[i * (VV + HH) + v] + Wh_b[i];
  float acc = Wd12_w[i * (VV + LATD) + v] + Wd12_b[i];
  for (int j = 0; j < LATD; j++) acc += Wd12_w[i * (VV + LATD) + VV + j] * latent[j];
  d12c[a * HH + i] = acc;
}

__global__ __launch_bounds__(512) void bias1_kernel(
    const float* Wl_w, const float* Wl_b, const float* latent, float* bias1) {
  int i = threadIdx.x;
  if (i >= HH) return;
  float acc = Wl_b[i];
  for (int j = 0; j < LATD; j++) acc += Wl_w[i * (HH + LATD) + j] * latent[j];
  bias1[i] = acc;
}

// column-major copies of the H x H recurrent blocks for coalesced matvecs
__global__ void transpose_kernel(const float* Wz_w, const float* Ur_w,
                                 const float* Wh_w, const float* Wd3_w,
                                 float* WzT, float* UrT, float* WhT, float* Wd3T) {
  int idx = blockIdx.x * blockDim.x + threadIdx.x;
  int stride = gridDim.x * blockDim.x;
  for (int t = idx; t < HH * HH; t += stride) {
    int i = t / HH, j = t % HH;
    WzT[j * HH + i]  = Wz_w[i * (VV + HH) + VV + j];
    UrT[j * HH + i]  = Ur_w[i * HH + j];
    WhT[j * HH + i]  = Wh_w[i * (VV + HH) + VV + j];
    Wd3T[j * HH + i] = Wd3_w[i * HH + j];
  }
}

// K-major zero-padded B operands for the WMMA GEMMs
__global__ void prepB_kernel(const float* Wl_w, const float* Ul_w, float* Bz1, float* B2) {
  int idx0 = blockIdx.x * blockDim.x + threadIdx.x;
  int stride = gridDim.x * blockDim.x;
  for (int t = idx0; t < KP * N1P; t += stride) {
    int k = t / N1P, n = t % N1P;
    Bz1[t] = (k < HH && n < HH) ? Wl_w[n * (HH + LATD) + LATD + k] : 0.f;
  }
  for (int t = idx0; t < KP * N2P; t += stride) {
    int k = t / N2P, n = t % N2P;
    B2[t] = (k < HH && n < VV) ? Ul_w[n * HH + k] : 0.f;
  }
}

// ------------------------------------------------------ sequential DFS scan
__global__ __launch_bounds__(512) void scan_kernel(
    const int* node_ids, const int* next_ids, const int* stops,
    const float* dz, const float* dr1, const float* dh, const float* d12c,
    const float* WzT, const float* UrT, const float* WhT, const float* Wd3T,
    const float* Wd3_b, const float* Ud_w, const float* Ud_b,
    float* msg, float* sumh, int* inc_count, int* inc_step,
    float* labH, float* out, int L) {
  __shared__ float ssum[HH], d3v[HH], svec[HH], mvec[HH], rsvec[HH];
  __shared__ float red[512];
  int tid = threadIdx.x;
  int labrow = 1;  // row 0 = root (zeros)
  for (int t = 0; t < L; t++) {
    int a = node_ids[t], b = next_ids[t], stop = stops[t];
    if (tid < HH) ssum[tid] = sumh[a * HH + tid];
    __syncthreads();
    // d3 = sumh[a] @ Wd3.T + b
    if (tid < HH) {
      float acc = Wd3_b[tid];
      for (int j = 0; j < HH; j++) acc += Wd3T[j * HH + tid] * ssum[j];
      d3v[tid] = acc;
    }
    __syncthreads();
    // pred_stop = relu([d12, d3]) . Ud + b
    float p = 0.f;
    for (int i = tid; i < 2 * HH; i += 512) {
      float v = (i < HH) ? d12c[a * HH + i] : d3v[i - HH];
      p += fmaxf(v, 0.f) * Ud_w[i];
    }
    red[tid] = p; __syncthreads();
    for (int off = 256; off > 0; off >>= 1) {
      if (tid < off) red[tid] += red[tid + off];
      __syncthreads();
    }
    if (tid == 0) { out[L + t] = red[0] + Ud_b[0]; out[t] = (float)stop; }
    // masked sum s and r-sum over active incoming messages (all except j==b)
    int cnt = inc_count[a];
    int K = 0;
    if (tid < HH) { svec[tid] = 0.f; rsvec[tid] = 0.f; }
    __syncthreads();
    for (int e = 0; e < cnt; e++) {
      int t2 = inc_step[a * NN + e];
      int j = node_ids[t2];
      if (j == b) continue;           // uniform branch
      K++;
      if (tid < HH) mvec[tid] = msg[(long)t2 * HH + tid];
      __syncthreads();
      if (tid < HH) {
        svec[tid] += mvec[tid];
        float y = dr1[a * HH + tid];
        for (int jj = 0; jj < HH; jj++) y += UrT[jj * HH + tid] * mvec[jj];
        rsvec[tid] += 1.f / (1.f + expf(-y));
      }
      __syncthreads();
    }
    if (tid < HH) {
      float r0 = 1.f / (1.f + expf(-dr1[a * HH + tid]));
      rsvec[tid] += (float)(NN - K) * r0;   // all zero-rows of Hm collapse
    }
    __syncthreads();
    // z, gated, new_h
    if (tid < HH) {
      float za = dz[a * HH + tid];
      float ga = dh[a * HH + tid];
      for (int j = 0; j < HH; j++) {
        za += WzT[j * HH + tid] * svec[j];
        ga += WhT[j * HH + tid] * rsvec[j];
      }
      float z = 1.f / (1.f + expf(-za));
      float g = tanhf(ga);
      float nh = (1.f - z) * svec[tid] + z * g;
      if (t < L - 1) {                       // is_last discards the update
        msg[(long)t * HH + tid] = nh;
        sumh[b * HH + tid] += nh;
      }
      if (stop == 1) labH[labrow * LW + tid] = nh;
    }
    if (stop == 1) labrow++;
    if (tid == 0 && t < L - 1) {
      int c = inc_count[b];
      inc_step[b * NN + c] = t;
      inc_count[b] = c + 1;
    }
    __syncthreads();
  }
}

// ------------------------------------------------------------- WMMA GEMMs
// D(16x16 tile) = A(M x Kp, row-major, K-padded) * B(Kp x Np, padded) + bias
__global__ __launch_bounds__(32) void wmma_gemm_kernel(
    const float* __restrict__ A, int lda,
    const float* __restrict__ B, int ldb,
    float* __restrict__ C, int ldc,
    int Kp, const float* __restrict__ bias, int nmax, int relu_mode) {
  int lane = threadIdx.x;
  int bx = blockIdx.x, by = blockIdx.y;
  int mrow = by * 16 + (lane & 15);
  int ncol = bx * 16 + (lane & 15);
  int koff = (lane >> 4) * 2;  // f32 16x16x4 A/B fragment layout
  v8f acc = {0.f, 0.f, 0.f, 0.f, 0.f, 0.f, 0.f, 0.f};
  for (int k0 = 0; k0 < Kp; k0 += 4) {
    v2f av, bv;
    av.x = A[mrow * lda + k0 + koff];
    av.y = A[mrow * lda + k0 + koff + 1];
    bv.x = B[(k0 + koff) * ldb + ncol];
    bv.y = B[(k0 + koff + 1) * ldb + ncol];
    acc = __builtin_amdgcn_wmma_f32_16x16x4_f32(
        false, av, false, bv, (short)0, acc, false, false);
  }
  if (ncol < nmax) {
    float bb = bias[ncol];
    for (int e = 0; e < 8; e++) {
      int row = by * 16 + e + 8 * (lane >> 4);
      float v = acc[e] + bb;
      if (relu_mode) v = fmaxf(v, 0.f);
      C[row * ldc + ncol] = v;
    }
  }
}

__global__ __launch_bounds__(256) void softmax_kernel(const float* logits, float* out, int base) {
  __shared__ float red[256];
  int r = blockIdx.x, tid = threadIdx.x;
  const float* row = logits + r * N2P;
  if (r == 0) {  // root label: raw logits, no softmax (matches reference)
    for (int i = tid; i < VV; i += 256) out[base + i] = row[i];
    return;
  }
  float m = -3.4e38f;
  for (int i = tid; i < VV; i += 256) m = fmaxf(m, row[i]);
  red[tid] = m; __syncthreads();
  for (int off = 128; off > 0; off >>= 1) {
    if (tid < off) red[tid] = fmaxf(red[tid], red[tid + off]);
    __syncthreads();
  }
  m = red[0]; __syncthreads();
  float s = 0.f;
  for (int i = tid; i < VV; i += 256) s += expf(row[i] - m);
  red[tid] = s; __syncthreads();
  for (int off = 128; off > 0; off >>= 1) {
    if (tid < off) red[tid] += red[tid + off];
    __syncthreads();
  }
  float inv = 1.f / red[0];
  for (int i = tid; i < VV; i += 256) out[base + r * VV + i] = expf(row[i] - m) * inv;
}

__global__ void copy_misc_kernel(const int* real_labels, float* out, int off, int n) {
  int i = blockIdx.x * blockDim.x + threadIdx.x;
  if (i < n) out[off + i] = (float)real_labels[i];
}

// ------------------------------------------------------------------ launch
extern "C" void kernel_launch(void* const* d_in, const int* in_sizes, int n_in,
                              void* d_out, int out_size, void* d_ws, size_t ws_size,
                              hipStream_t stream) {
  (void)n_in; (void)out_size; (void)ws_size;
  const float* latent       = (const float*)d_in[0];
  const float* node_feature = (const float*)d_in[1];
  const int*   node_ids     = (const int*)d_in[2];
  const int*   next_ids     = (const int*)d_in[3];
  const int*   stops        = (const int*)d_in[4];
  const int*   real_labels  = (const int*)d_in[6];
  const float* Wz_w  = (const float*)d_in[7];  const float* Wz_b  = (const float*)d_in[8];
  const float* Ur_w  = (const float*)d_in[9];
  const float* Wr_w  = (const float*)d_in[10]; const float* Wr_b  = (const float*)d_in[11];
  const float* Wh_w  = (const float*)d_in[12]; const float* Wh_b  = (const float*)d_in[13];
  const float* Wd12_w= (const float*)d_in[14]; const float* Wd12_b= (const float*)d_in[15];
  const float* Wd3_w = (const float*)d_in[16]; const float* Wd3_b = (const float*)d_in[17];
  const float* Ud_w  = (const float*)d_in[18]; const float* Ud_b  = (const float*)d_in[19];
  const float* Wl_w  = (const float*)d_in[20]; const float* Wl_b  = (const float*)d_in[21];
  const float* Ul_w  = (const float*)d_in[22]; const float* Ul_b  = (const float*)d_in[23];
  float* out = (float*)d_out;
  int L    = in_sizes[2];
  int NLAB = in_sizes[6];

  char* p = (char*)d_ws;
  auto carve = [&](size_t bytes) -> void* {
    void* q = (void*)p;
    p += (bytes + 255) & ~(size_t)255;
    return q;
  };
  int* vid       = (int*)carve(NN * sizeof(int));
  int* inc_step  = (int*)carve((size_t)NN * NN * sizeof(int));
  // contiguous zero-init region: inc_count, sumh, labH, A2
  char* zbeg = p;
  int*   inc_count = (int*)carve(NN * sizeof(int));
  float* sumh  = (float*)carve((size_t)NN * HH * sizeof(float));
  float* labH  = (float*)carve((size_t)NN * LW * sizeof(float));
  float* A2    = (float*)carve((size_t)NN * LW * sizeof(float));
  char* zend = p;
  float* dz    = (float*)carve((size_t)NN * HH * sizeof(float));
  float* dr1   = (float*)carve((size_t)NN * HH * sizeof(float));
  float* dh    = (float*)carve((size_t)NN * HH * sizeof(float));
  float* d12c  = (float*)carve((size_t)NN * HH * sizeof(float));
  float* bias1 = (float*)carve(HH * sizeof(float));
  float* WzT   = (float*)carve((size_t)HH * HH * sizeof(float));
  float* UrT   = (float*)carve((size_t)HH * HH * sizeof(float));
  float* WhT   = (float*)carve((size_t)HH * HH * sizeof(float));
  float* Wd3T  = (float*)carve((size_t)HH * HH * sizeof(float));
  float* msg   = (float*)carve((size_t)L * HH * sizeof(float));
  float* Bz1   = (float*)carve((size_t)KP * N1P * sizeof(float));
  float* B2    = (float*)carve((size_t)KP * N2P * sizeof(float));
  float* logits= (float*)carve((size_t)NN * N2P * sizeof(float));

  long zero_words = (long)(zend - zbeg) / 4;
  zero_kernel<<<1024, 256, 0, stream>>>((float*)zbeg, zero_words);
  vid_kernel<<<NN, 128, 0, stream>>>(node_feature, vid);
  pernode_kernel<<<NN, 512, 0, stream>>>(vid, latent, Wz_w, Wz_b, Wr_w, Wr_b,
                                         Wh_w, Wh_b, Wd12_w, Wd12_b,
                                         dz, dr1, dh, d12c);
  bias1_kernel<<<1, 512, 0, stream>>>(Wl_w, Wl_b, latent, bias1);
  transpose_kernel<<<512, 256, 0, stream>>>(Wz_w, Ur_w, Wh_w, Wd3_w, WzT, UrT, WhT, Wd3T);
  prepB_kernel<<<512, 256, 0, stream>>>(Wl_w, Ul_w, Bz1, B2);
  scan_kernel<<<1, 512, 0, stream>>>(node_ids, next_ids, stops,
                                     dz, dr1, dh, d12c, WzT, UrT, WhT, Wd3T,
                                     Wd3_b, Ud_w, Ud_b,
                                     msg, sumh, inc_count, inc_step,
                                     labH, out, L);
  copy_misc_kernel<<<(NLAB + 255) / 256, 256, 0, stream>>>(real_labels, out, 2 * L, NLAB);
  // t1 = relu(labH @ WlH.T + bias1)            (256 x 450), WMMA f32
  wmma_gemm_kernel<<<dim3(N1P / 16, NN / 16), 32, 0, stream>>>(
      labH, LW, Bz1, N1P, A2, LW, KP, bias1, HH, 1);
  // logits = t1 @ Ul.T + Ul_b                  (256 x 780), WMMA f32
  wmma_gemm_kernel<<<dim3(N2P / 16, NN / 16), 32, 0, stream>>>(
      A2, LW, B2, N2P, logits, N2P, KP, Ul_b, VV, 0);
  softmax_kernel<<<NLAB, 256, 0, stream>>>(logits, out, 2 * L + NLAB);
}